// Linear_42262478192871
// MI455X (gfx1250) — compile-verified
//
#include <hip/hip_runtime.h>

// CDNA5 / gfx1250: wave32, WMMA 16x16 shapes. Full-fp32 GEMM via V_WMMA_F32_16X16X4_F32.
// Staging via GLOBAL_LOAD_ASYNC_TO_LDS_B128 (ASYNCcnt). k-outer loop: B fragments are
// transient (4 VGPRs), only accumulators stay resident.

typedef float v2f __attribute__((ext_vector_type(2)));
typedef float v8f __attribute__((ext_vector_type(8)));
typedef int   v4i __attribute__((ext_vector_type(4)));

#if defined(__has_builtin)
#if __has_builtin(__builtin_amdgcn_global_load_async_to_lds_b128)
#define USE_ASYNC_LDS 1
#endif
#endif
#ifndef USE_ASYNC_LDS
#define USE_ASYNC_LDS 0
#endif

typedef __attribute__((address_space(1))) v4i* g4_ptr;
typedef __attribute__((address_space(3))) v4i* l4_ptr;

constexpr float K_ALPHA = 0.08838834764831845f; // 128^-0.5
constexpr int ROW = 1152; // floats per node row (128*1 + 128*3 + 128*5)

// Compute NT row tiles x 2 col tiles for one wave. k outer, tiles inner:
// per k-step -> 2 transient B frags + NT A frags + 2*NT independent WMMAs.
template <int D, int CW, int NT, int XOFF, int WOFF>
__device__ __forceinline__ void gemm_tiles(const float* __restrict__ sx,
                                           const float* __restrict__ wt,
                                           float* __restrict__ y,
                                           int z0, int tstart,
                                           int n, int kh, int mhi, int col0, int col1) {
    // LDS base offset of each tile's A row for this lane (element u lives at +u*D)
    int aoff[NT];
#pragma unroll
    for (int tt = 0; tt < NT; ++tt) {
        const int r  = (tstart + tt) * 16 + n;
        const int zl = r / D;
        aoff[tt] = zl * CW + (r - zl * D);
    }

    const v8f vzero = {0.f, 0.f, 0.f, 0.f, 0.f, 0.f, 0.f, 0.f};
    v8f acc0[NT], acc1[NT];
#pragma unroll
    for (int tt = 0; tt < NT; ++tt) { acc0[tt] = vzero; acc1[tt] = vzero; }

    const float* wb = wt + WOFF;
#pragma unroll
    for (int k = 0; k < 32; ++k) {
        const int K = 4 * k + kh;
        // B 4x16 fp32 layout: V0 = {lanes0-15: K, lanes16-31: K+2}, V1 = {K+1, K+3}.
        v2f bb0, bb1;
        bb0.x = wb[K * 128 + col0];
        bb0.y = wb[(K + 1) * 128 + col0];
        bb1.x = wb[K * 128 + col1];
        bb1.y = wb[(K + 1) * 128 + col1];
#pragma unroll
        for (int tt = 0; tt < NT; ++tt) {
            v2f af;
            af.x = sx[aoff[tt] + K * D];
            af.y = sx[aoff[tt] + (K + 1) * D];
            // 8 args: (neg_a, A, neg_b, B, c_mod, C, reuse_a, reuse_b)
            acc0[tt] = __builtin_amdgcn_wmma_f32_16x16x4_f32(
                false, af, false, bb0, (short)0, acc0[tt], false, false);
            acc1[tt] = __builtin_amdgcn_wmma_f32_16x16x4_f32(
                false, af, false, bb1, (short)0, acc1[tt], false, false);
        }
    }

    // D layout: VGPR j holds row M=j (lanes 0-15) / M=j+8 (lanes 16-31), col N = n.
#pragma unroll
    for (int tt = 0; tt < NT; ++tt) {
#pragma unroll
        for (int j = 0; j < 8; ++j) {
            const int rl = (tstart + tt) * 16 + j + mhi;
            const int zz = rl / D;
            const int i2 = rl - zz * D;
            const size_t base = (size_t)(z0 + zz) * ROW + XOFF + i2;
            __builtin_nontemporal_store(K_ALPHA * acc0[tt][j], &y[base + col0 * D]);
            __builtin_nontemporal_store(K_ALPHA * acc1[tt][j], &y[base + col1 * D]);
        }
    }
}

// One irrep block: y[z, XOFF + w*D + i] = alpha * sum_u W[WOFF + u*128 + w] * x[z, XOFF + u*D + i]
// 8 waves = 4 column-pairs (32 cols each) x 2 row-groups.
template <int D, int ZC, int XOFF, int WOFF>
__global__ __launch_bounds__(256) void irrep_linear_kernel(const float* __restrict__ x,
                                                           const float* __restrict__ wt,
                                                           float* __restrict__ y) {
    constexpr int CW  = 128 * D;        // floats per node in this block slice
    constexpr int T   = (ZC * D) / 16;  // 16-row tiles per chunk
    constexpr int NT0 = (T + 1) / 2;    // tiles owned by row-group 0
    constexpr int NT1 = T - NT0;        // tiles owned by row-group 1
    __shared__ float sx[ZC * CW];

    const int z0  = blockIdx.x * ZC;
    const int tid = threadIdx.x;

    // ---- Stage x chunk slice into LDS (async B128 copies; rows are contiguous) ----
    constexpr int NV4 = (ZC * CW) / 4;
    for (int v = tid; v < NV4; v += 256) {
        const int z = v / (CW / 4);
        const int c = v - z * (CW / 4);
        const float* gsrc = x + (size_t)(z0 + z) * ROW + XOFF + c * 4;
        float* ldst = &sx[z * CW + c * 4];
#if USE_ASYNC_LDS
        __builtin_amdgcn_global_load_async_to_lds_b128((g4_ptr)gsrc, (l4_ptr)ldst,
                                                       /*offset=*/0, /*cpol=*/0);
#else
        *reinterpret_cast<float4*>(ldst) = *reinterpret_cast<const float4*>(gsrc);
#endif
    }
#if USE_ASYNC_LDS
    asm volatile("s_wait_asynccnt 0x0" ::: "memory");
#endif
    __syncthreads();

    const int lane    = tid & 31;
    const int n       = lane & 15;          // N within tile / M within tile (A)
    const int kh      = (lane >> 4) << 1;   // K offset: 0 (lanes 0-15) or 2 (lanes 16-31)
    const int mhi     = (lane >> 4) << 3;   // +8 row offset for D-matrix high lanes
    const int colpair = (tid >> 5) & 3;     // which 32-column group
    const int rowgrp  = (tid >> 5) >> 2;    // 0 or 1
    const int col0    = colpair * 32 + n;
    const int col1    = col0 + 16;

    if constexpr (NT0 == NT1) {
        gemm_tiles<D, CW, NT0, XOFF, WOFF>(sx, wt, y, z0, rowgrp * NT0,
                                           n, kh, mhi, col0, col1);
    } else {
        // Wave-uniform branch (rowgrp is constant across the wave): each side
        // executes with EXEC all-ones, or is branched over entirely.
        if (rowgrp == 0) {
            gemm_tiles<D, CW, NT0, XOFF, WOFF>(sx, wt, y, z0, 0,
                                               n, kh, mhi, col0, col1);
        } else {
            gemm_tiles<D, CW, NT1, XOFF, WOFF>(sx, wt, y, z0, NT0,
                                               n, kh, mhi, col0, col1);
        }
    }
}

extern "C" void kernel_launch(void* const* d_in, const int* in_sizes, int n_in,
                              void* d_out, int out_size, void* d_ws, size_t ws_size,
                              hipStream_t stream) {
    (void)in_sizes; (void)n_in; (void)d_ws; (void)ws_size; (void)out_size;
    const float* x = (const float*)d_in[0];
    const float* w = (const float*)d_in[1];
    float* y = (float*)d_out;

    // N_NODES = 100000; chunk sizes chosen so 100000 % ZC == 0 and ZC*D % 16 == 0,
    // and static LDS (ZC*128*D*4B) stays <= 64KB: 16KB / 48KB / 40KB.
    irrep_linear_kernel<1, 32,   0,     0><<<dim3(100000 / 32), dim3(256), 0, stream>>>(x, w, y);
    irrep_linear_kernel<3, 32, 128, 16384><<<dim3(100000 / 32), dim3(256), 0, stream>>>(x, w, y);
    irrep_linear_kernel<5, 16, 512, 32768><<<dim3(100000 / 16), dim3(256), 0, stream>>>(x, w, y);
}